// T5GNNAdapt_80444737454183
// MI455X (gfx1250) — compile-verified
//
#include <hip/hip_runtime.h>
#include <hip/hip_bf16.h>
#include <math.h>

// ---------------------------------------------------------------------------
// T5 GNN adapter layer for gfx1250 (MI455X).
//   out = hidden + ELU( RGCN(RMSNorm(hidden)) ) @ wo
// RGCN restructured: per-edge matmul pushed through segment_sum ->
//   deterministic bucketed aggregation (no fp atomics) + fp32 WMMA GEMMs.
// ---------------------------------------------------------------------------

#define D_MODEL 1024
#define D_FF    286
#define D_FF_P  288      // padded to multiple of 16
#define K_TOT   3072     // D_MODEL * (1 root + 2 relations)
#define CHUNK   2048     // LDS bucket chunk (8 KB)

typedef __attribute__((ext_vector_type(2))) float v2f;
typedef __attribute__((ext_vector_type(8))) float v8f;

static __device__ __forceinline__ v8f wmma_f32_k4(v2f a, v2f b, v8f c) {
  // V_WMMA_F32_16X16X4_F32 : D = A(16x4) * B(4x16) + C, exact fp32
  return __builtin_amdgcn_wmma_f32_16x16x4_f32(
      /*neg_a=*/false, a, /*neg_b=*/false, b,
      /*c_mod=*/(short)0, c, /*reuse_a=*/false, /*reuse_b=*/false);
}

// ---------------------------------------------------------------------------
// Kernel: pack weights.
//   Wp  [288 cols][3072 k]  column-major combined [W_root; W_rel0; W_rel1]
//   WOp [1024 cols][288 k]  column-major wo_weight, K padded 286->288 w/ 0
// ---------------------------------------------------------------------------
__global__ __launch_bounds__(256)
void pack_weights_kernel(const float* __restrict__ W_rel,
                         const float* __restrict__ W_root,
                         const float* __restrict__ wo,
                         float* __restrict__ Wp,
                         float* __restrict__ WOp) {
  const int total1 = D_FF_P * K_TOT;          // 884736
  const int total2 = D_MODEL * D_FF_P;        // 294912
  int idx = blockIdx.x * 256 + threadIdx.x;
  if (idx < total1) {
    int n = idx / K_TOT;
    int k = idx - n * K_TOT;
    float v = 0.0f;
    if (n < D_FF) {
      int region = k >> 10;          // 0: root, 1: rel0, 2: rel1
      int kk = k & 1023;
      if (region == 0) v = W_root[kk * D_FF + n];
      else             v = W_rel[((region - 1) * D_MODEL + kk) * D_FF + n];
    }
    Wp[n * K_TOT + k] = v;
  } else if (idx < total1 + total2) {
    int j = idx - total1;
    int n = j / D_FF_P;
    int k = j - n * D_FF_P;
    WOp[n * D_FF_P + k] = (k < D_FF) ? wo[k * D_MODEL + n] : 0.0f;
  }
}

// ---------------------------------------------------------------------------
// Kernel: T5 RMSNorm. One wave (32 lanes) per row, 8 rows per 256-thr block.
// ---------------------------------------------------------------------------
__global__ __launch_bounds__(256)
void rmsnorm_kernel(const float* __restrict__ x,
                    const float* __restrict__ lnw,
                    float* __restrict__ normx, int N) {
  int wave = threadIdx.x >> 5;
  int lane = threadIdx.x & 31;
  int row  = blockIdx.x * 8 + wave;
  if (row >= N) return;
  const float4* rp = (const float4*)(x + (size_t)row * D_MODEL);
  float4 xv[8];
  float ss = 0.0f;
#pragma unroll
  for (int i = 0; i < 8; ++i) {
    xv[i] = rp[i * 32 + lane];
    ss += xv[i].x * xv[i].x + xv[i].y * xv[i].y +
          xv[i].z * xv[i].z + xv[i].w * xv[i].w;
  }
#pragma unroll
  for (int off = 16; off > 0; off >>= 1) ss += __shfl_xor(ss, off, 32);
  float scale = rsqrtf(ss * (1.0f / (float)D_MODEL) + 1e-6f);
  float4* op = (float4*)(normx + (size_t)row * D_MODEL);
  const float4* wp = (const float4*)lnw;
#pragma unroll
  for (int i = 0; i < 8; ++i) {
    float4 w = wp[i * 32 + lane];
    float4 o;
    o.x = xv[i].x * scale * w.x;  o.y = xv[i].y * scale * w.y;
    o.z = xv[i].z * scale * w.z;  o.w = xv[i].w * scale * w.w;
    op[i * 32 + lane] = o;
  }
}

// ---------------------------------------------------------------------------
// Kernel: histogram of (type,dst) -> cnt32[2N]  (int atomics only)
// ---------------------------------------------------------------------------
__global__ __launch_bounds__(256)
void histogram_kernel(const int* __restrict__ eidx, const int* __restrict__ etype,
                      int* __restrict__ cnt32, int N, int E) {
  int e = blockIdx.x * 256 + threadIdx.x;
  if (e >= E) return;
  int dst = eidx[E + e];
  int t   = etype[e];
  atomicAdd(&cnt32[t * N + dst], 1);
}

// ---------------------------------------------------------------------------
// Kernel: exclusive scan over 2N bins (single block, 1024 threads).
// Writes offs[] and a working copy cursor[].
// ---------------------------------------------------------------------------
__global__ __launch_bounds__(1024)
void scan_kernel(const int* __restrict__ cnt32, int* __restrict__ offs,
                 int* __restrict__ cursor, int B) {
  __shared__ int part[1024];
  int tid = threadIdx.x;
  int chunk = (B + 1023) >> 10;
  int base = tid * chunk;
  int s = 0;
  for (int j = 0; j < chunk; ++j) {
    int i = base + j;
    if (i < B) s += cnt32[i];
  }
  part[tid] = s;
  __syncthreads();
  // Hillis-Steele inclusive scan
  for (int off = 1; off < 1024; off <<= 1) {
    int v = part[tid];
    int add = (tid >= off) ? part[tid - off] : 0;
    __syncthreads();
    part[tid] = v + add;
    __syncthreads();
  }
  int run = part[tid] - s;   // exclusive prefix for this thread's chunk
  for (int j = 0; j < chunk; ++j) {
    int i = base + j;
    if (i < B) {
      offs[i]   = run;
      cursor[i] = run;
      run += cnt32[i];
    }
  }
}

// ---------------------------------------------------------------------------
// Kernel: bucket placement. Each edge claims a slot in its (type,dst) bucket
// and stores its src node. Slot order is racy; fixed by sort in accumulate.
// ---------------------------------------------------------------------------
__global__ __launch_bounds__(256)
void placement_kernel(const int* __restrict__ eidx, const int* __restrict__ etype,
                      int* __restrict__ cursor, int* __restrict__ bucketSrc,
                      int N, int E) {
  int e = blockIdx.x * 256 + threadIdx.x;
  if (e >= E) return;
  int src = eidx[e];
  int dst = eidx[E + e];
  int t   = etype[e];
  int pos = atomicAdd(&cursor[t * N + dst], 1);
  bucketSrc[pos] = src;
}

// ---------------------------------------------------------------------------
// Kernel: deterministic per-(type,dst) accumulation.
//   agg[t][dst][:] = ( sum over sorted bucket of norm_x[src][:] ) / max(cnt,1)
// One 256-thread block per bucket; plain coalesced float4 reads (no fp atomics).
// Also zero-fills rows with empty buckets (replaces memset + scale pass).
// ---------------------------------------------------------------------------
__global__ __launch_bounds__(256)
void accumulate_kernel(const int* __restrict__ cnt32, const int* __restrict__ offs,
                       const int* __restrict__ bucketSrc,
                       const float* __restrict__ normx,
                       float* __restrict__ Xbase, int N) {
  __shared__ int srcs[CHUNK];
  int b = blockIdx.x;              // 0 .. 2N-1 ;  t = b/N, dst = b%N
  int tid = threadIdx.x;
  int count = cnt32[b];
  int start = offs[b];
  float4 acc = make_float4(0.f, 0.f, 0.f, 0.f);
  for (int done = 0; done < count; done += CHUNK) {
    int c = min(count - done, CHUNK);
    for (int i = tid; i < c; i += 256) srcs[i] = bucketSrc[start + done + i];
    __syncthreads();
    if (tid == 0) {                 // insertion sort -> deterministic fp order
      for (int i = 1; i < c; ++i) {
        int v = srcs[i], j = i - 1;
        while (j >= 0 && srcs[j] > v) { srcs[j + 1] = srcs[j]; --j; }
        srcs[j + 1] = v;
      }
    }
    __syncthreads();
    for (int i = 0; i < c; ++i) {
      const float4* rp = (const float4*)(normx + (size_t)srcs[i] * D_MODEL);
      float4 v = rp[tid];
      acc.x += v.x; acc.y += v.y; acc.z += v.z; acc.w += v.w;
    }
    __syncthreads();
  }
  float s = 1.0f / fmaxf((float)count, 1.0f);
  int t = (b >= N) ? 1 : 0;
  int dst = b - t * N;
  float4* op = (float4*)(Xbase + ((size_t)(1 + t) * N + dst) * D_MODEL);
  op[tid] = make_float4(acc.x * s, acc.y * s, acc.z * s, acc.w * s);
}

// ---------------------------------------------------------------------------
// GEMM1: ff[N x 288] = ELU( X[N x 3072] @ Wp + bias )
// One wave per 16-row x 96-col strip (6 WMMA tiles). grid = (N/16, 3).
// ---------------------------------------------------------------------------
__global__ __launch_bounds__(32)
void gemm_ffn_kernel(const float* __restrict__ X,     // 3 stacked [N,1024] blocks
                     const float* __restrict__ Wp,    // col-major [288][3072]
                     const float* __restrict__ bias,  // [286]
                     float* __restrict__ ff, int N) {
  const int lane = threadIdx.x;
  const int m    = lane & 15;
  const int half = lane >> 4;
  const int row  = blockIdx.x * 16 + m;
  const int colBase = blockIdx.y * 96;

  v8f acc[6];
#pragma unroll
  for (int t = 0; t < 6; ++t) acc[t] = (v8f)0.0f;

  const float* ap = X + (size_t)row * D_MODEL + 2 * half;
  const float* bp[6];
#pragma unroll
  for (int t = 0; t < 6; ++t)
    bp[t] = Wp + (size_t)(colBase + t * 16 + m) * K_TOT + 2 * half;

  const size_t regStride = (size_t)N * D_MODEL;
  for (int k = 0; k < K_TOT; k += 4) {
    int region = k >> 10;
    int kk = k & 1023;
    v2f a = *(const v2f*)(ap + (size_t)region * regStride + kk);
#pragma unroll
    for (int t = 0; t < 6; ++t) {
      v2f b = *(const v2f*)(bp[t] + k);
      acc[t] = wmma_f32_k4(a, b, acc[t]);
    }
  }

  // epilogue: + bias, ELU, store (C layout: vgpr i -> M = i + 8*half, N = m)
#pragma unroll
  for (int t = 0; t < 6; ++t) {
    int cc = colBase + t * 16 + m;
    float bv = (cc < D_FF) ? bias[cc] : 0.0f;
#pragma unroll
    for (int i = 0; i < 8; ++i) {
      int rr = blockIdx.x * 16 + half * 8 + i;
      float v = acc[t][i] + bv;
      float e = (v > 0.0f) ? v : (__expf(v) - 1.0f);
      ff[(size_t)rr * D_FF_P + cc] = e;
    }
  }
}

// ---------------------------------------------------------------------------
// GEMM2: out[N x 1024] = hidden + ff[N x 288] @ WOp
// One wave per 16-row x 128-col strip (8 WMMA tiles). grid = (N/16, 8).
// ---------------------------------------------------------------------------
__global__ __launch_bounds__(32)
void gemm_out_kernel(const float* __restrict__ ff,
                     const float* __restrict__ WOp,   // col-major [1024][288]
                     const float* __restrict__ hidden,
                     float* __restrict__ out, int N) {
  const int lane = threadIdx.x;
  const int m    = lane & 15;
  const int half = lane >> 4;
  const int row  = blockIdx.x * 16 + m;
  const int colBase = blockIdx.y * 128;

  v8f acc[8];
#pragma unroll
  for (int t = 0; t < 8; ++t) acc[t] = (v8f)0.0f;

  const float* ap = ff + (size_t)row * D_FF_P + 2 * half;
  const float* bp[8];
#pragma unroll
  for (int t = 0; t < 8; ++t)
    bp[t] = WOp + (size_t)(colBase + t * 16 + m) * D_FF_P + 2 * half;

  for (int k = 0; k < D_FF_P; k += 4) {
    v2f a = *(const v2f*)(ap + k);
#pragma unroll
    for (int t = 0; t < 8; ++t) {
      v2f b = *(const v2f*)(bp[t] + k);
      acc[t] = wmma_f32_k4(a, b, acc[t]);
    }
  }

#pragma unroll
  for (int t = 0; t < 8; ++t) {
    int cc = colBase + t * 16 + m;
#pragma unroll
    for (int i = 0; i < 8; ++i) {
      int rr = blockIdx.x * 16 + half * 8 + i;
      size_t o = (size_t)rr * D_MODEL + cc;
      out[o] = hidden[o] + acc[t][i];
    }
  }
}

// ---------------------------------------------------------------------------
extern "C" void kernel_launch(void* const* d_in, const int* in_sizes, int n_in,
                              void* d_out, int out_size, void* d_ws, size_t ws_size,
                              hipStream_t stream) {
  const float* hidden  = (const float*)d_in[0];
  const int*   eidx    = (const int*)d_in[1];
  const int*   etype   = (const int*)d_in[2];
  const float* lnw     = (const float*)d_in[3];
  const float* W_rel   = (const float*)d_in[4];
  const float* W_root  = (const float*)d_in[5];
  const float* bias    = (const float*)d_in[6];
  const float* wo      = (const float*)d_in[7];
  float*       out     = (float*)d_out;

  const int N = in_sizes[0] / D_MODEL;   // 20000
  const int E = in_sizes[2];             // 320000
  const int B = 2 * N;                   // buckets = NUM_REL * N

  // workspace layout
  float* X    = (float*)d_ws;                       // [3][N][1024]: norm|agg0|agg1
  float* ff   = X + (size_t)3 * N * D_MODEL;        // [N][288]
  float* Wp   = ff + (size_t)N * D_FF_P;            // [288][3072] col-major
  float* WOp  = Wp + (size_t)D_FF_P * K_TOT;        // [1024][288] col-major
  int* cnt32     = (int*)(WOp + (size_t)D_MODEL * D_FF_P);  // [2N]
  int* offs      = cnt32 + B;                               // [2N]
  int* cursor    = offs + B;                                // [2N]
  int* bucketSrc = cursor + B;                              // [E]

  hipMemsetAsync(cnt32, 0, (size_t)B * sizeof(int), stream);

  {
    int total = D_FF_P * K_TOT + D_MODEL * D_FF_P;
    pack_weights_kernel<<<(total + 255) / 256, 256, 0, stream>>>(
        W_rel, W_root, wo, Wp, WOp);
  }

  rmsnorm_kernel<<<(N + 7) / 8, 256, 0, stream>>>(hidden, lnw, X, N);

  histogram_kernel<<<(E + 255) / 256, 256, 0, stream>>>(eidx, etype, cnt32, N, E);

  scan_kernel<<<1, 1024, 0, stream>>>(cnt32, offs, cursor, B);

  placement_kernel<<<(E + 255) / 256, 256, 0, stream>>>(eidx, etype, cursor,
                                                        bucketSrc, N, E);

  accumulate_kernel<<<B, 256, 0, stream>>>(cnt32, offs, bucketSrc, X, X, N);

  gemm_ffn_kernel<<<dim3((N + 15) / 16, 3), 32, 0, stream>>>(X, Wp, bias, ff, N);

  gemm_out_kernel<<<dim3((N + 15) / 16, 8), 32, 0, stream>>>(ff, WOp, hidden, out, N);
}